// ClustGeoEdgeEncoder_72060961292777
// MI455X (gfx1250) — compile-verified
//
#include <hip/hip_runtime.h>

typedef __attribute__((ext_vector_type(2))) float v2f;
typedef __attribute__((ext_vector_type(8))) float v8f;

#define P 128
#define TILE 16

// One block (4 wave32s) per edge. LDS-staged clusters, V_WMMA_F32_16X16X4_F32
// builds the 128x128 distance matrix (64 tiles, 16 per wave). Branchless
// argmin via 64-bit packed (value_bits, flat_idx) keys + u64 min, reduced
// with __shfl_xor within each wave and a tiny LDS combine across waves.
__launch_bounds__(128)
__global__ void ClustGeoEdgeEncoder_kernel(const float* __restrict__ data,
                                           const int* __restrict__ clusts,
                                           const int* __restrict__ clust_sizes,
                                           const int* __restrict__ edge_index,
                                           float* __restrict__ out,
                                           int E)
{
    __shared__ __align__(16) float ldsA[P * 4];          // (x, y, z, 1)
    __shared__ __align__(16) float ldsB[P * 4];          // (-2x', -2y', -2z', sq2')
    __shared__ __align__(16) float ldsS[P];              // sq1' (1e30 if invalid)
    __shared__ unsigned long long redK[4];               // per-wave argmin keys

    const int e = blockIdx.x;
    const int t = threadIdx.x;
    const int src = edge_index[e];
    const int dst = edge_index[E + e];

    // ---- stage both clusters into LDS (coords are data columns 1..3) ----
    {
        const int p  = t;
        const int ia = clusts[src * P + p];
        const float ax = data[ia * 8 + 1];
        const float ay = data[ia * 8 + 2];
        const float az = data[ia * 8 + 3];
        ldsA[p * 4 + 0] = ax;
        ldsA[p * 4 + 1] = ay;
        ldsA[p * 4 + 2] = az;
        ldsA[p * 4 + 3] = 1.0f;
        ldsS[p] = (p < clust_sizes[src]) ? (ax * ax + ay * ay + az * az) : 1e30f;

        const int ib = clusts[dst * P + p];
        const float bx = data[ib * 8 + 1];
        const float by = data[ib * 8 + 2];
        const float bz = data[ib * 8 + 3];
        ldsB[p * 4 + 0] = -2.0f * bx;
        ldsB[p * 4 + 1] = -2.0f * by;
        ldsB[p * 4 + 2] = -2.0f * bz;
        ldsB[p * 4 + 3] = (p < clust_sizes[dst]) ? (bx * bx + by * by + bz * bz) : 1e30f;
    }
    __syncthreads();

    const int lane = t & 31;
    const int wave = t >> 5;
    const int hi = lane >> 4;   // selects K pair {0,1} vs {2,3}
    const int lo = lane & 15;   // row (A) / col (B) within 16-wide tile

    const v2f*    pA = (const v2f*)ldsA;
    const v2f*    pB = (const v2f*)ldsB;
    const float4* pS = (const float4*)ldsS;

    unsigned long long best = ~0ull;

    // 8 row-tiles total; each wave owns 2 row-tiles x 8 col-tiles = 16 WMMAs.
    #pragma unroll
    for (int r = 0; r < 2; ++r) {
        const int m0 = (wave * 2 + r) * TILE;
        const v2f a = pA[(m0 + lo) * 2 + hi];              // A fragment (2 VGPRs)
        const int sb = (m0 + 8 * hi) >> 2;                 // sq1 for my 8 rows
        const float4 s0 = pS[sb];
        const float4 s1 = pS[sb + 1];
        const float sq[8] = {s0.x, s0.y, s0.z, s0.w, s1.x, s1.y, s1.z, s1.w};
        const int mbase = (m0 + 8 * hi) * P + lo;

        #pragma unroll
        for (int ct = 0; ct < 8; ++ct) {
            const int n0 = ct * TILE;
            const v2f b = pB[(n0 + lo) * 2 + hi];          // B fragment (2 VGPRs)
            v8f c = {};
            // c = A(16x4) * B(4x16) = -2*inner + sq2'
            c = __builtin_amdgcn_wmma_f32_16x16x4_f32(
                    false, a, false, b, (short)0, c, false, false);
            #pragma unroll
            for (int v = 0; v < 8; ++v) {
                const float d = fmaxf(c[v] + sq[v], 0.0f);     // d >= 0
                const unsigned int idx = (unsigned int)(mbase + v * P + n0);
                const unsigned long long key =
                    ((unsigned long long)__float_as_uint(d) << 32) | idx;
                best = (key < best) ? key : best;              // branchless
            }
        }
    }

    // ---- wave32 argmin reduction (branchless, 5 shuffle steps) ----
    #pragma unroll
    for (int off = 16; off > 0; off >>= 1) {
        const unsigned long long other = __shfl_xor(best, off, 32);
        best = (other < best) ? other : best;
    }
    if (lane == 0) redK[wave] = best;
    __syncthreads();

    if (t == 0) {
        unsigned long long bk = redK[0];
        bk = (redK[1] < bk) ? redK[1] : bk;
        bk = (redK[2] < bk) ? redK[2] : bk;
        bk = (redK[3] < bk) ? redK[3] : bk;
        const unsigned int bi = (unsigned int)(bk & 0xffffffffu);
        const int i1 = (int)(bi >> 7);      // / P
        const int i2 = (int)(bi & (P - 1));
        const float v1x = ldsA[i1 * 4 + 0];
        const float v1y = ldsA[i1 * 4 + 1];
        const float v1z = ldsA[i1 * 4 + 2];
        const float v2x = -0.5f * ldsB[i2 * 4 + 0];
        const float v2y = -0.5f * ldsB[i2 * 4 + 1];
        const float v2z = -0.5f * ldsB[i2 * 4 + 2];
        float dx = v1x - v2x, dy = v1y - v2y, dz = v1z - v2z;
        const float lend = sqrtf(dx * dx + dy * dy + dz * dz);
        if (lend > 0.0f) {
            dx /= lend; dy /= lend; dz /= lend;
        }
        float* o = out + (size_t)e * 19;
        o[0]  = v1x; o[1]  = v1y; o[2]  = v1z;
        o[3]  = v2x; o[4]  = v2y; o[5]  = v2z;
        o[6]  = dx;  o[7]  = dy;  o[8]  = dz;
        o[9]  = lend;
        o[10] = dx * dx; o[11] = dx * dy; o[12] = dx * dz;
        o[13] = dy * dx; o[14] = dy * dy; o[15] = dy * dz;
        o[16] = dz * dx; o[17] = dz * dy; o[18] = dz * dz;
    }
}

extern "C" void kernel_launch(void* const* d_in, const int* in_sizes, int n_in,
                              void* d_out, int out_size, void* d_ws, size_t ws_size,
                              hipStream_t stream) {
    const float* data        = (const float*)d_in[0];
    const int*   clusts      = (const int*)d_in[1];
    const int*   clust_sizes = (const int*)d_in[2];
    const int*   edge_index  = (const int*)d_in[3];
    float*       out         = (float*)d_out;

    const int E = in_sizes[3] / 2;   // edge_index is (2, E)

    ClustGeoEdgeEncoder_kernel<<<E, 128, 0, stream>>>(
        data, clusts, clust_sizes, edge_index, out, E);
}